// NearestEmbedEMA_52432960750159
// MI455X (gfx1250) — compile-verified
//
#include <hip/hip_runtime.h>
#include <hip/hip_bf16.h>

typedef __attribute__((ext_vector_type(2))) float v2f;
typedef __attribute__((ext_vector_type(8))) float v8f;

#define DIMD 64
#define HWSZ 1024
#define NTOK 131072
#define NEMB 512
#define DECAYF 0.99f
#define ONEMD 0.01f
#define EPSV 1e-5f

// ws layout (floats / ints): [0,512) counts(int), [512, 33280) embed_sum(float), [33280, 33792) wnorm(float)
#define WS_COUNTS 0
#define WS_ESUM 512
#define WS_WNORM 33280

// out layout (floats), concatenated in reference return order
#define OUT_RESULT 0
#define OUT_ARGMIN 8388608
#define OUT_WEIGHT 8519680
#define OUT_CSIZE  8552448
#define OUT_EAVG   8552960
#define OUT_PREV   8585728

__global__ void k_zero(float* ws) {
    int i = blockIdx.x * 256 + threadIdx.x;
    if (i < WS_WNORM) ws[i] = 0.0f; // zeros counts (int 0) + embed_sum
}

__global__ void k_wnorm(const float* __restrict__ w, float* __restrict__ ws) {
    int k = blockIdx.x * 256 + threadIdx.x;
    if (k < NEMB) {
        float s = 0.0f;
        #pragma unroll 8
        for (int d = 0; d < DIMD; ++d) { float v = w[d * NEMB + k]; s += v * v; }
        ws[WS_WNORM + k] = s;
    }
}

__launch_bounds__(256)
__global__ void k_main(const float* __restrict__ x, const float* __restrict__ w,
                       const float* __restrict__ ws, float* __restrict__ out) {
    extern __shared__ float smem[];
    float* sw   = smem;                               // 32768: codebook (D x NEMB)
    float* swn  = smem + DIMD * NEMB;                 // 512:  ||w_k||^2
    int*   amem = (int*)(smem + DIMD * NEMB + NEMB);  // 256: per-wave argmin staging

    const int tid = threadIdx.x;
    for (int i = tid; i < DIMD * NEMB; i += 256) sw[i] = w[i];
    for (int i = tid; i < NEMB; i += 256) swn[i] = ws[WS_WNORM + i];
    __syncthreads();

    const int lane = tid & 31;
    const int wv   = tid >> 5;
    const int half = lane >> 4;
    const int l15  = lane & 15;
    const int koff = half * 2;

    const int t0 = (blockIdx.x * 8 + wv) * 32;        // 32 tokens per wave (2 M-tiles)

    // A fragments (16x4 f32 layout): lane = M (token), {reg, lane-half} = K (dim). A = -2*x.
    const int tok0 = t0 + l15;
    const int tok1 = t0 + 16 + l15;
    const float* xb0 = x + (size_t)(tok0 >> 10) * DIMD * HWSZ + (tok0 & (HWSZ - 1));
    const float* xb1 = x + (size_t)(tok1 >> 10) * DIMD * HWSZ + (tok1 & (HWSZ - 1));
    v2f a0[16], a1[16];
    #pragma unroll
    for (int kc = 0; kc < 16; ++kc) {
        int d0 = kc * 4 + koff;
        a0[kc].x = -2.0f * xb0[d0 * HWSZ];
        a0[kc].y = -2.0f * xb0[(d0 + 1) * HWSZ];
        a1[kc].x = -2.0f * xb1[d0 * HWSZ];
        a1[kc].y = -2.0f * xb1[(d0 + 1) * HWSZ];
    }

    float mv0[8], mv1[8];
    int   mi0[8], mi1[8];
    #pragma unroll
    for (int r = 0; r < 8; ++r) {
        mv0[r] = 3.4e38f; mi0[r] = 0;
        mv1[r] = 3.4e38f; mi1[r] = 0;
    }

    // B-fragment loader: tile n covers codes [n*16, n*16+16)
    auto load_b = [&](v2f* bb, int n) {
        const int code = n * 16 + l15;
        #pragma unroll
        for (int kc = 0; kc < 16; ++kc) {
            int d0 = kc * 4 + koff;
            bb[kc].x = sw[d0 * NEMB + code];
            bb[kc].y = sw[(d0 + 1) * NEMB + code];
        }
    };
    // Two independent accumulation chains sharing one B buffer
    auto compute = [&](const v2f* bb, int n) {
        const int code = n * 16 + l15;
        const float wn = swn[code];
        v8f c0 = {};   // inline-0 SRC2 for first WMMA; ||w||^2 folded in at compare time
        v8f c1 = {};
        #pragma unroll
        for (int kc = 0; kc < 16; ++kc) {
            c0 = __builtin_amdgcn_wmma_f32_16x16x4_f32(false, a0[kc], false, bb[kc],
                                                       (short)0, c0, false, false);
            c1 = __builtin_amdgcn_wmma_f32_16x16x4_f32(false, a1[kc], false, bb[kc],
                                                       (short)0, c1, false, false);
        }
        #pragma unroll
        for (int r = 0; r < 8; ++r) {
            float s0 = c0[r] + wn;
            float s1 = c1[r] + wn;
            if (s0 < mv0[r]) { mv0[r] = s0; mi0[r] = code; } // tiles ascend -> lowest idx on ties
            if (s1 < mv1[r]) { mv1[r] = s1; mi1[r] = code; }
        }
    };

    v2f bbA[16], bbB[16];
    load_b(bbA, 0);
    for (int n = 0; n < NEMB / 16; n += 2) {
        load_b(bbB, n + 1);                 // prefetch while bbA computes
        compute(bbA, n);
        if (n + 2 < NEMB / 16) load_b(bbA, n + 2);
        compute(bbB, n + 1);
    }

    // Min-reduce across the 16 lanes of each half (C: lanes0-15 M=r, lanes16-31 M=r+8)
    #pragma unroll
    for (int m = 1; m <= 8; m <<= 1) {
        #pragma unroll
        for (int r = 0; r < 8; ++r) {
            float ov0 = __shfl_xor(mv0[r], m, 32);
            int   oi0 = __shfl_xor(mi0[r], m, 32);
            if (ov0 < mv0[r] || (ov0 == mv0[r] && oi0 < mi0[r])) { mv0[r] = ov0; mi0[r] = oi0; }
            float ov1 = __shfl_xor(mv1[r], m, 32);
            int   oi1 = __shfl_xor(mi1[r], m, 32);
            if (ov1 < mv1[r] || (ov1 == mv1[r] && oi1 < mi1[r])) { mv1[r] = ov1; mi1[r] = oi1; }
        }
    }

    if (l15 == 0) {
        #pragma unroll
        for (int r = 0; r < 8; ++r) {
            amem[wv * 32 + half * 8 + r]      = mi0[r];
            amem[wv * 32 + 16 + half * 8 + r] = mi1[r];
        }
    }
    __syncthreads();

    // Emit argmin (as float) + quantized result gathered from LDS codebook
    #pragma unroll
    for (int g = 0; g < 2; ++g) {
        const int tok = t0 + g * 16 + l15;
        const int k = amem[wv * 32 + g * 16 + l15];
        if (half == 0) out[OUT_ARGMIN + tok] = (float)k;
        const int b = tok >> 10;
        const int hw = tok & (HWSZ - 1);
        float* ob = out + OUT_RESULT + (size_t)b * DIMD * HWSZ + hw;
        #pragma unroll 8
        for (int dd = 0; dd < 32; ++dd) {
            int d = half * 32 + dd;
            ob[d * HWSZ] = sw[d * NEMB + k];
        }
    }
}

__launch_bounds__(256)
__global__ void k_scatter(const float* __restrict__ x, const float* __restrict__ out,
                          float* __restrict__ ws) {
    extern __shared__ float smem[];
    int*   lcnt = (int*)smem;        // 512
    float* lsum = smem + NEMB;       // 32768
    const int tid = threadIdx.x;
    for (int i = tid; i < NEMB + DIMD * NEMB; i += 256) smem[i] = 0.0f;
    __syncthreads();

    for (int t = blockIdx.x * 256 + tid; t < NTOK; t += gridDim.x * 256) {
        const int k = (int)out[OUT_ARGMIN + t];
        atomicAdd(&lcnt[k], 1);
        const int b = t >> 10;
        const int hw = t & (HWSZ - 1);
        const float* xb = x + (size_t)b * DIMD * HWSZ + hw;
        #pragma unroll 8
        for (int d = 0; d < DIMD; ++d) {
            atomicAdd(&lsum[d * NEMB + k], xb[d * HWSZ]);
        }
    }
    __syncthreads();

    int* gcnt = (int*)ws;
    for (int i = tid; i < NEMB; i += 256) atomicAdd(&gcnt[i], lcnt[i]);
    for (int i = tid; i < DIMD * NEMB; i += 256) atomicAdd(&ws[WS_ESUM + i], lsum[i]);
}

__launch_bounds__(512)
__global__ void k_final(const float* __restrict__ ws,
                        const float* __restrict__ cluster_size,
                        const float* __restrict__ embed_avg,
                        const float* __restrict__ prev_cluster,
                        float* __restrict__ out) {
    __shared__ float red[NEMB];
    const int k = threadIdx.x;
    const float cnt = (float)((const int*)ws)[WS_COUNTS + k];
    const float cnt_c = (cnt == 0.0f) ? 1.0f : cnt;
    const float ncs = cluster_size[k] * DECAYF + ONEMD * cnt_c;
    out[OUT_CSIZE + k] = ncs;
    out[OUT_PREV + k] = prev_cluster[k] + cnt;
    red[k] = ncs;
    __syncthreads();
    for (int s = 256; s > 0; s >>= 1) {
        if (k < s) red[k] += red[k + s];
        __syncthreads();
    }
    const float n = red[0];
    const float inv = 1.0f / ((ncs + EPSV) / (n + NEMB * EPSV) * n);
    for (int d = 0; d < DIMD; ++d) {
        const float ea = embed_avg[d * NEMB + k] * DECAYF + ONEMD * ws[WS_ESUM + d * NEMB + k];
        out[OUT_EAVG + d * NEMB + k] = ea;
        out[OUT_WEIGHT + d * NEMB + k] = ea * inv;
    }
}

extern "C" void kernel_launch(void* const* d_in, const int* in_sizes, int n_in,
                              void* d_out, int out_size, void* d_ws, size_t ws_size,
                              hipStream_t stream) {
    const float* x    = (const float*)d_in[0];
    const float* w    = (const float*)d_in[1];
    const float* csz  = (const float*)d_in[2];
    const float* eavg = (const float*)d_in[3];
    const float* prev = (const float*)d_in[4];
    float* out = (float*)d_out;
    float* ws  = (float*)d_ws;

    k_zero<<<(WS_WNORM + 255) / 256, 256, 0, stream>>>(ws);
    k_wnorm<<<2, 256, 0, stream>>>(w, ws);

    const size_t lds_main = (size_t)(DIMD * NEMB + NEMB + 256) * 4; // ~134 KB (<=320 KB/WGP)
    k_main<<<NTOK / 256, 256, lds_main, stream>>>(x, w, ws, out);

    const size_t lds_sc = (size_t)(NEMB + DIMD * NEMB) * 4;         // ~133 KB
    k_scatter<<<32, 256, lds_sc, stream>>>(x, out, ws);

    k_final<<<1, 512, 0, stream>>>(ws, csz, eavg, prev, out);
}